// ModelNew_73315091744563
// MI455X (gfx1250) — compile-verified
//
#include <hip/hip_runtime.h>
#include <hip/hip_bf16.h>

typedef float v2f __attribute__((ext_vector_type(2)));
typedef float v8f __attribute__((ext_vector_type(8)));

// Problem constants
#define NB     32          // batch
#define CIN    3
#define COUT   16
#define XD     32          // input spatial dim
#define XSP    (XD*XD*XD)  // 32768 elems per (n,ci)
#define OD     63          // conv_transpose output spatial dim
#define PD     15          // pooled output dim
#define POUT   (NB*COUT*PD*PD*PD)   // 1,728,000
#define NTAP   28                   // 27 real taps + 1 zero slot
#define TASKS  (NB*OD*OD*4)         // 508,032 wave-tasks
#define WAVES_PER_BLOCK 8
#define BLOCKS1 (TASKS/WAVES_PER_BLOCK)  // 63,504

__global__ void zero_ws_kernel(float* __restrict__ p, int n) {
    int i = blockIdx.x * blockDim.x + threadIdx.x;
    if (i < n) p[i] = 0.0f;
}

// Pass 1: fused transposed-conv (f32 WMMA) + channel sum/sumsq + 4x4x4 block sums.
// One wave = 16 output voxels (strided w-line of uniform parity) x 16 output channels.
// Fixed 8-tap unrolled loop; duplicated taps for even-parity dims hit a zero-weight slot.
__global__ void __launch_bounds__(256)
convt_stats_pool_kernel(const float* __restrict__ x,
                        const float* __restrict__ wt,
                        double* __restrict__ gsum,   // [16]
                        double* __restrict__ gsq,    // [16]
                        float*  __restrict__ P)      // [32,16,15,15,15] block sums
{
    // packed B operands: sWp[(co*28 + tap)*4 + ci] = W[ci][co][tap]; tap 27 == zeros
    __shared__ __align__(16) float sWp[COUT * NTAP * 4];
    __shared__ float sSum[COUT];
    __shared__ float sSq[COUT];

    for (int i = threadIdx.x; i < COUT * NTAP; i += blockDim.x) {
        const int co = i / NTAP, tap = i % NTAP;
        float w0 = 0.f, w1 = 0.f, w2 = 0.f;
        if (tap < 27) {
            w0 = wt[      co * 27 + tap];
            w1 = wt[432 + co * 27 + tap];
            w2 = wt[864 + co * 27 + tap];
        }
        sWp[i * 4 + 0] = w0; sWp[i * 4 + 1] = w1;
        sWp[i * 4 + 2] = w2; sWp[i * 4 + 3] = 0.f;
    }
    if (threadIdx.x < COUT) { sSum[threadIdx.x] = 0.0f; sSq[threadIdx.x] = 0.0f; }
    __syncthreads();

    const int lane = threadIdx.x & 31;
    const int wv   = threadIdx.x >> 5;

    // decode task = ((n*63 + od)*63 + oh)*4 + tile
    long task = (long)blockIdx.x * WAVES_PER_BLOCK + wv;
    const int tile = (int)(task & 3);  task >>= 2;
    const int oh   = (int)(task % OD); task /= OD;
    const int od   = (int)(task % OD); task /= OD;
    const int n    = (int)task;
    const int p    = tile & 1;   // w parity of this tile
    const int tw   = tile >> 1;  // which half of the w-line

    const int hi = lane >> 4;    // lane half: ci {0,1} vs {2,pad}
    const int m  = lane & 15;    // A: voxel index; B/D: output channel

    const int ow_m  = p + tw * 32 + 2 * m;   // this lane's output w (A voxel)
    const bool vld  = (ow_m < OD);           // only m=15,p=1,tw=1 is out of range
    const float* xn = x + (size_t)n * (CIN * XSP);
    const float* xA = xn + (hi ? (size_t)(2 * XSP) : 0);  // base for A.x (ci0 / ci2)

    // fixed-length tap lists; z==1 marks the duplicated (zero-weight) tap
    int kda[2], zda[2], kha[2], zha[2], kwa[2], zwa[2];
    if (od & 1) { kda[0]=0; kda[1]=2; zda[0]=0; zda[1]=0; }
    else        { kda[0]=1; kda[1]=1; zda[0]=0; zda[1]=1; }
    if (oh & 1) { kha[0]=0; kha[1]=2; zha[0]=0; zha[1]=0; }
    else        { kha[0]=1; kha[1]=1; zha[0]=0; zha[1]=1; }
    if (p)      { kwa[0]=0; kwa[1]=2; zwa[0]=0; zwa[1]=0; }
    else        { kwa[0]=1; kwa[1]=1; zwa[0]=0; zwa[1]=1; }

    v8f acc = {0.f,0.f,0.f,0.f,0.f,0.f,0.f,0.f};

#pragma unroll
    for (int ia = 0; ia < 2; ++ia) {
        const int id = (od + 1 - kda[ia]) >> 1;
#pragma unroll
        for (int ib = 0; ib < 2; ++ib) {
            const int ih = (oh + 1 - kha[ib]) >> 1;
#pragma unroll
            for (int ic = 0; ic < 2; ++ic) {
                int iw = (ow_m + 1 - kwa[ic]) >> 1;
                if (iw > XD - 1) iw = XD - 1;            // safe addr for masked lane
                const size_t sx = (size_t)id * (XD*XD) + (size_t)ih * XD + iw;

                // unconditional loads (addresses always in range), lane-masked by select
                const float ax = xA[sx];                 // ci0 (lo) / ci2 (hi)
                const float ay = xn[sx + XSP];           // ci1 (valid addr for all lanes)
                v2f A;
                A.x = vld ? ax : 0.0f;
                A.y = (vld && !hi) ? ay : 0.0f;

                // one ds_load_b64: this lane's half of the 4x16 B operand
                const int z   = zda[ia] | zha[ib] | zwa[ic];
                const int tap = z ? 27 : (kda[ia]*9 + kha[ib]*3 + kwa[ic]);
                const v2f B = *(const v2f*)&sWp[(m * NTAP + tap) * 4 + hi * 2];

                acc = __builtin_amdgcn_wmma_f32_16x16x4_f32(
                        false, A, false, B, (short)0, acc, false, false);
            }
        }
    }

    // D layout: lane reg r holds y[voxel r + 8*hi][co = m]
    // (1) channel statistics: reduce 8 regs, fold lane<->lane+16, LDS accumulate
    float s = 0.0f, q = 0.0f;
#pragma unroll
    for (int r = 0; r < 8; ++r) { const float v = acc[r]; s += v; q += v * v; }
    s += __shfl_xor(s, 16, 32);
    q += __shfl_xor(q, 16, 32);
    if (lane < 16) { atomicAdd(&sSum[m], s); atomicAdd(&sSq[m], q); }

    // (2) pooled 4x4x4 block sums (positions >= 60 fall off the double pool)
    if ((od < 60) & (oh < 60)) {
        const size_t pbase =
            ((((size_t)n * COUT + m) * PD + (od >> 2)) * PD + (oh >> 2)) * PD;
#pragma unroll
        for (int r = 0; r < 8; ++r) {
            const int vm = r + 8 * hi;
            const int ow = p + tw * 32 + 2 * vm;
            if (ow < 60) atomicAdd(&P[pbase + (ow >> 2)], acc[r]);
        }
    }

    __syncthreads();
    if (threadIdx.x < COUT) {
        atomicAdd(&gsum[threadIdx.x], (double)sSum[threadIdx.x]);
        atomicAdd(&gsq[threadIdx.x],  (double)sSq[threadIdx.x]);
    }
}

// Pass 2: out = (P/64 - mean) * rsqrt(var+eps) * gamma + beta
__global__ void finalize_kernel(const float* __restrict__ P,
                                const double* __restrict__ gsum,
                                const double* __restrict__ gsq,
                                const float* __restrict__ gamma,
                                const float* __restrict__ beta,
                                float* __restrict__ out)
{
    const int i = blockIdx.x * blockDim.x + threadIdx.x;
    if (i >= POUT) return;
    const int c = (i / (PD*PD*PD)) % COUT;
    const double cnt  = (double)NB * OD * OD * OD;
    const double mean = gsum[c] / cnt;
    const double var  = gsq[c] / cnt - mean * mean;
    const float scale = rsqrtf((float)var + 1e-5f) * gamma[c];
    out[i] = (P[i] * (1.0f / 64.0f) - (float)mean) * scale + beta[c];
}

extern "C" void kernel_launch(void* const* d_in, const int* in_sizes, int n_in,
                              void* d_out, int out_size, void* d_ws, size_t ws_size,
                              hipStream_t stream) {
    const float* x     = (const float*)d_in[0];
    const float* wt    = (const float*)d_in[1];
    const float* gamma = (const float*)d_in[2];
    const float* beta  = (const float*)d_in[3];
    float* out = (float*)d_out;

    // workspace layout: [ 16 x f64 sum | 16 x f64 sumsq | P: 1,728,000 x f32 ]
    double* gsum = (double*)d_ws;
    double* gsq  = gsum + 16;
    float*  P    = (float*)((char*)d_ws + 256);

    const int nzero = 64 + POUT;  // 32 doubles viewed as 64 floats, then P
    zero_ws_kernel<<<(nzero + 255) / 256, 256, 0, stream>>>((float*)d_ws, nzero);

    convt_stats_pool_kernel<<<BLOCKS1, 256, 0, stream>>>(x, wt, gsum, gsq, P);

    finalize_kernel<<<(POUT + 255) / 256, 256, 0, stream>>>(P, gsum, gsq, gamma, beta, out);
}